// kernel_generated_0_62904091017319
// MI455X (gfx1250) — compile-verified
//
#include <hip/hip_runtime.h>

typedef float v2f __attribute__((ext_vector_type(2)));
typedef float v4f __attribute__((ext_vector_type(4)));
typedef float v8f __attribute__((ext_vector_type(8)));
typedef int   v4i __attribute__((vector_size(4 * sizeof(int))));  // matches builtin

#define Hdim 56
#define Wdim 56
#define Cdim 128
#define Odim 32
#define KWdim 3
#define KDIM (Cdim * KWdim)            // 384
#define XSTRIDE 64                     // x-row stride in LDS (56 data + 8 zero halo)
#define GUARD 16                       // zeroed guard floats before xbuf0 (left edge c=0)
#define XBUF_FLOATS (Cdim * XSTRIDE)   // 8192 floats = 32 KB per buffer
#define LDS_W_FLOATS (Odim * KDIM)     // 12288 floats = 48 KB
#define HCHUNK 8

#define AS1 __attribute__((address_space(1)))
#define AS3 __attribute__((address_space(3)))

#if defined(__has_builtin)
#if __has_builtin(__builtin_amdgcn_global_load_async_to_lds_b128)
#define USE_ASYNC_LDS 1
#endif
#endif

__device__ __forceinline__ void async_fence_lds() {
#if defined(USE_ASYNC_LDS)
#if __has_builtin(__builtin_amdgcn_s_wait_asynccnt)
  __builtin_amdgcn_s_wait_asynccnt(0);
#else
  asm volatile("s_wait_asynccnt 0" ::: "memory");
#endif
#endif
}

// Stage x[n, :, h, :] (128 ch x 56 floats) into LDS buffer [c][0..55].
// 14 x b128 per channel, 16-byte aligned on both sides.
__device__ __forceinline__ void stage_row(float* dst, const float* src, int tid) {
  for (int i = tid; i < Cdim * (Wdim / 4); i += 128) {  // 1792 b128 transfers
    int c = i / 14, q = i - c * 14;
    const float* g = src + (size_t)c * (Hdim * Wdim) + q * 4;
    float* l = dst + c * XSTRIDE + q * 4;
#if defined(USE_ASYNC_LDS)
    __builtin_amdgcn_global_load_async_to_lds_b128((AS1 v4i*)g, (AS3 v4i*)l,
                                                   0, 0);
#else
    *(v4f*)l = *(const v4f*)g;
#endif
  }
}

// D = A(16x4,f32) x B(4x16,f32) + C, wave32 WMMA. One block = 4 waves; wave i
// owns output columns [16i,16i+16) of one (n,h) row, computing all 32 output
// channels with two accumulators sharing each B fragment. K = kw*128 + c.
// x rows are double-buffered: async-load row h+1 while WMMAs chew on row h.
extern "C" __global__ __launch_bounds__(128)
void conv1x3_wmma_f32(const float* __restrict__ xg,
                      const float* __restrict__ wg,
                      float* __restrict__ yg) {
  extern __shared__ float lds[];
  float* xbuf0 = lds + GUARD;
  float* xbuf1 = xbuf0 + XBUF_FLOATS;
  float* wlds  = xbuf1 + XBUF_FLOATS;   // [O][KDIM], K permuted to kw-major

  const int tid  = threadIdx.x;
  const int lane = tid & 31;
  const int wave = tid >> 5;            // 0..3 -> W tile
  const int half = lane >> 4;           // lane half selects K sub-pair
  const int hw   = lane & 15;           // M row (A/D) or N col (B/D)
  const int n    = blockIdx.y;
  const int h0   = blockIdx.x * HCHUNK;

  const float* xbase = xg + (size_t)n * Cdim * (Hdim * Wdim);

  // ---- stage weights once per block, permuted so K = kw*128 + c ----
  for (int i = tid; i < LDS_W_FLOATS; i += 128) {
    int o  = i / KDIM;
    int k  = i - o * KDIM;
    int kw = k >> 7;
    int c  = k & 127;
    wlds[i] = wg[(o * Cdim + c) * KWdim + kw];
  }
  // ---- zero guard row and both buffers' halo columns (written once) ----
  for (int i = tid; i < GUARD; i += 128) lds[i] = 0.0f;
  for (int i = tid; i < 2 * Cdim * 8; i += 128) {
    int b = i >> 10;                    // buffer 0/1
    int t = i & 1023;
    int c = t >> 3, j = (t & 7) + 56;   // halo columns 56..63
    (b ? xbuf1 : xbuf0)[c * XSTRIDE + j] = 0.0f;
  }
  // ---- kick row h0 into buffer 0, then close the prologue ----
  stage_row(xbuf0, xbase + (size_t)h0 * Wdim, tid);
  async_fence_lds();
  __syncthreads();

  const int w0 = wave * 16;
  const float* arow0 = &wlds[hw * KDIM];           // A rows for M = hw
  const float* arow1 = &wlds[(hw + 16) * KDIM];    // A rows for M = hw + 16

  for (int hh = 0; hh < HCHUNK; ++hh) {
    const int h = h0 + hh;
    const float* xcur = (hh & 1) ? xbuf1 : xbuf0;
    float* xnxt = (hh & 1) ? xbuf0 : xbuf1;

    // ---- overlap: async-stage row h+1 while computing row h ----
    if (hh + 1 < HCHUNK)
      stage_row(xnxt, xbase + (size_t)(h + 1) * Wdim, tid);

    // ---- GEMM: 96 K-steps, 2 WMMAs per step sharing one B fragment ----
    v8f acc0 = {};   // output channels 0..15
    v8f acc1 = {};   // output channels 16..31
    for (int kw = 0; kw < KWdim; ++kw) {
      const int bofs = w0 + hw + kw - 1;   // window column; -1/56+ hit zeros
#pragma unroll 4
      for (int c0 = 0; c0 < Cdim; c0 += 4) {
        const int kl = (kw << 7) + c0 + half * 2;   // this lane's K pair
        v2f a0 = *(const v2f*)(arow0 + kl);
        v2f a1 = *(const v2f*)(arow1 + kl);
        const int cb = c0 + half * 2;               // B rows K -> channels
        v2f b;
        b.x = xcur[cb * XSTRIDE + bofs];
        b.y = xcur[(cb + 1) * XSTRIDE + bofs];
        acc0 = __builtin_amdgcn_wmma_f32_16x16x4_f32(
            false, a0, false, b, (short)0, acc0, false, false);
        acc1 = __builtin_amdgcn_wmma_f32_16x16x4_f32(
            false, a1, false, b, (short)0, acc1, false, false);
      }
    }

    // ---- store with roll(+1) in H; mask columns >= 56 (tile 3 overhang) ----
    int oh = h + 1; if (oh >= Hdim) oh -= Hdim;
    const int wcol = w0 + hw;
    if (wcol < Wdim) {
      size_t base = (size_t)n * Odim * (Hdim * Wdim) + oh * Wdim + wcol;
#pragma unroll
      for (int v = 0; v < 8; ++v) {
        const int m = v + half * 8;                 // D layout: M = v + 8*half
        yg[base + (size_t)m * (Hdim * Wdim)]        = acc0[v];
        yg[base + (size_t)(m + 16) * (Hdim * Wdim)] = acc1[v];
      }
    }

    // next buffer must be fully staged (all waves) before anyone reads it
    async_fence_lds();
    __syncthreads();
  }
}

extern "C" void kernel_launch(void* const* d_in, const int* in_sizes, int n_in,
                              void* d_out, int out_size, void* d_ws, size_t ws_size,
                              hipStream_t stream) {
  (void)in_sizes; (void)n_in; (void)out_size; (void)d_ws; (void)ws_size;
  const float* x = (const float*)d_in[0];
  const float* w = (const float*)d_in[1];
  float* y = (float*)d_out;

  dim3 grid(Hdim / HCHUNK, 128);   // (7, 128): 8 h-rows per block, n per y
  dim3 block(128);                 // 4 wave32s
  size_t shmem =
      (size_t)(GUARD + 2 * XBUF_FLOATS + LDS_W_FLOATS) * sizeof(float); // ~113 KB
  hipLaunchKernelGGL(conv1x3_wmma_f32, grid, block, shmem, stream, x, w, y);
}